// ROI_46042049413475
// MI455X (gfx1250) — compile-verified
//
#include <hip/hip_runtime.h>
#include <stdint.h>

// ---------------------------------------------------------------------------
// ROI max pooling (torchvision-style, matches the JAX reference bit-for-bit
// in the fp32 bin arithmetic).
//
// features: (B=2, C=128, H=50, W=50) f32     rois: (N=256, 5) f32
// out:      (N, C, 7, 7) f32
//
// CDNA5 usage: GLOBAL_LOAD_ASYNC_TO_LDS_B32 (ASYNCcnt-tracked DMA into LDS,
// bypassing VGPRs) + s_wait_asynccnt before the workgroup barrier. wave32,
// 256-thread blocks = 8 waves.
// ---------------------------------------------------------------------------

#define ROI_B 2
#define ROI_C 128
#define ROI_H 50
#define ROI_W 50
#define OUT_HW 7
#define CCH 4                 // channels staged per block
#define LDS_ELEMS (CCH * ROI_H * ROI_W)   // 10000 floats = 40 KB

#if __has_builtin(__builtin_amdgcn_global_load_async_to_lds_b32)
#define HAVE_ASYNC_BUILTIN 1
#else
#define HAVE_ASYNC_BUILTIN 0
#endif

typedef __attribute__((address_space(1))) int GInt;   // global (AS1)
typedef __attribute__((address_space(3))) int LInt;   // LDS    (AS3)

__device__ __forceinline__ int iclip(int v, int lo, int hi) {
    return v < lo ? lo : (v > hi ? hi : v);
}

__device__ __forceinline__ void async_wait_all() {
#if __has_builtin(__builtin_amdgcn_s_wait_asynccnt)
    __builtin_amdgcn_s_wait_asynccnt(0);
#else
    asm volatile("s_wait_asynccnt 0x0" ::: "memory");
#endif
}

__device__ __forceinline__ void async_g2lds_b32(const float* g, float* l) {
#if HAVE_ASYNC_BUILTIN
    GInt* gp = (GInt*)const_cast<float*>(g);  // generic -> AS1, typed int*
    LInt* lp = (LInt*)l;                      // generic -> AS3, typed int*
    __builtin_amdgcn_global_load_async_to_lds_b32(gp, lp, 0, 0);
#else
    uint32_t lofs = (uint32_t)(uintptr_t)(LInt*)l;         // LDS offset (AS3)
    uint64_t ga   = (uint64_t)(uintptr_t)g;                // flat == global VA
    asm volatile("global_load_async_to_lds_b32 %0, %1, off"
                 :: "v"(lofs), "v"(ga) : "memory");
#endif
}

__global__ __launch_bounds__(256)
void roi_max_pool_kernel(const float* __restrict__ feat,
                         const float* __restrict__ rois,
                         float* __restrict__ out) {
    __shared__ float lds[LDS_ELEMS];

    const int blk = blockIdx.x;
    const int roi = blk >> 5;           // / (C/CCH) == /32
    const int c0  = (blk & 31) * CCH;
    const int tid = threadIdx.x;

    // ---- decode ROI (redundantly per thread; scalar-broadcast loads) ----
    const float* r = rois + (size_t)roi * 5;
    const int b  = (int)r[0];
    const int x1 = (int)rintf(r[1] * 0.0625f);   // jnp.round == RNE
    const int y1 = (int)rintf(r[2] * 0.0625f);
    const int x2 = (int)rintf(r[3] * 0.0625f);
    const int y2 = (int)rintf(r[4] * 0.0625f);

    const int rw_i = (x2 - x1 + 1) > 1 ? (x2 - x1 + 1) : 1;
    const int rh_i = (y2 - y1 + 1) > 1 ? (y2 - y1 + 1) : 1;
    const float bin_w = (float)rw_i / 7.0f;
    const float bin_h = (float)rh_i / 7.0f;

    // Bounding window = union of all bins (pw=0 start .. pw=6 end, same fp ops)
    const int wlo = iclip(x1, 0, ROI_W);
    const int whi = iclip((int)ceilf(7.0f * bin_w) + x1, 0, ROI_W);
    const int hlo = iclip(y1, 0, ROI_H);
    const int hhi = iclip((int)ceilf(7.0f * bin_h) + y1, 0, ROI_H);
    const int Wwin = whi - wlo;
    const int Hwin = hhi - hlo;

    // ---- stage CCH channel windows into LDS via async DMA ----
    if (Wwin > 0 && Hwin > 0) {
        const int plane = Hwin * Wwin;
        const int nelem = CCH * plane;
        const float* base = feat + ((size_t)b * ROI_C + c0) * (ROI_H * ROI_W);
        for (int i = tid; i < nelem; i += 256) {
            const int cl  = i / plane;
            const int rem = i - cl * plane;
            const int hh  = rem / Wwin;
            const int ww  = rem - hh * Wwin;
            const float* g = base + cl * (ROI_H * ROI_W)
                                  + (hlo + hh) * ROI_W + (wlo + ww);
            async_g2lds_b32(g, &lds[i]);
        }
    }
    async_wait_all();
    __syncthreads();

    // ---- each of 196 threads computes one (channel, bin) output ----
    if (tid < CCH * OUT_HW * OUT_HW) {
        const int cl  = tid / (OUT_HW * OUT_HW);
        const int bin = tid - cl * (OUT_HW * OUT_HW);
        const int ph  = bin / OUT_HW;
        const int pw  = bin - ph * OUT_HW;
        const float phf = (float)ph, pwf = (float)pw;

        const int hs = iclip((int)floorf(phf * bin_h) + y1, 0, ROI_H);
        const int he = iclip((int)ceilf((phf + 1.0f) * bin_h) + y1, 0, ROI_H);
        const int ws = iclip((int)floorf(pwf * bin_w) + x1, 0, ROI_W);
        const int we = iclip((int)ceilf((pwf + 1.0f) * bin_w) + x1, 0, ROI_W);

        float v = 0.0f;
        if (he > hs && we > ws) {
            float m = -3.402823466e38f;   // -FLT_MAX == finfo(f32).min
            const float* p = lds + cl * (Hwin * Wwin);
            for (int hh = hs; hh < he; ++hh) {
                const float* row = p + (hh - hlo) * Wwin - wlo;
                for (int ww = ws; ww < we; ++ww)
                    m = fmaxf(m, row[ww]);
            }
            v = m;   // non-empty bin of finite data can never stay at -FLT_MAX
        }
        out[(((size_t)roi * ROI_C + (c0 + cl)) * OUT_HW + ph) * OUT_HW + pw] = v;
    }
}

extern "C" void kernel_launch(void* const* d_in, const int* in_sizes, int n_in,
                              void* d_out, int out_size, void* d_ws, size_t ws_size,
                              hipStream_t stream) {
    const float* feat = (const float*)d_in[0];
    const float* rois = (const float*)d_in[1];
    float* out = (float*)d_out;
    const int N = in_sizes[1] / 5;                 // 256 ROIs
    const int blocks = N * (ROI_C / CCH);          // 8192
    roi_max_pool_kernel<<<blocks, 256, 0, stream>>>(feat, rois, out);
}